// CosL2DAM_70738111365250
// MI455X (gfx1250) — compile-verified
//
#include <hip/hip_runtime.h>
#include <hip/hip_bf16.h>
#include <math.h>

// ---------------- problem constants (from reference) ----------------
#define B_     256
#define D_     128
#define M_     16384
#define N_     20000
#define DEPTH_ 50

__device__ __constant__ const float SQRT_BETA = 2.0f;        // sqrt(4.0)
#define SQ2M   0.011048543456039806f                          // sqrt(2/M)
#define ALPHA_ 0.1f
#define EPS_   1e-5f
#define NTILES ((N_) / 16)                                    // 1250

typedef float v2f __attribute__((ext_vector_type(2)));
typedef float v8f __attribute__((ext_vector_type(8)));

// ======================================================================
// init: zero workspace accumulators (T, e_acc, g_acc) and copy q -> x
// ws layout (floats): T[M_], e_acc[B_], g_acc[B_*D_]
// ======================================================================
#define WS_FLOATS (M_ + B_ + B_ * D_)

__global__ __launch_bounds__(256) void dam_init_kernel(
    const float* __restrict__ q, float* __restrict__ x, float* __restrict__ wsf) {
  int i = blockIdx.x * 256 + threadIdx.x;
  if (i < B_ * D_) x[i] = q[i];
  if (i < WS_FLOATS) wsf[i] = 0.0f;
}

// ======================================================================
// memT: T[m] = sqrt(2/M) * sum_n cos( 2 * (memories[n] . S[m]) + bias[m] )
// block = 256 threads = 8 waves; wave w owns m-cols [m0, m0+16)
// grid.y strides over the 1250 n-tiles of 16 rows each.
// ======================================================================
__global__ __launch_bounds__(256) void dam_memT_kernel(
    const float* __restrict__ mem, const float* __restrict__ S,
    const float* __restrict__ bias, float* __restrict__ T) {
  const int tid  = threadIdx.x;
  const int wave = tid >> 5;
  const int lane = tid & 31;
  const int l    = lane & 15;
  const int half = lane >> 4;
  const int m0   = blockIdx.x * 128 + wave * 16;

  const float* srow = S + (size_t)(m0 + l) * D_ + 2 * half;  // B-matrix row
  const float  bia  = bias[m0 + l];

  float colsum = 0.0f;
  for (int t = blockIdx.y; t < NTILES; t += gridDim.y) {
    const int    n0   = t * 16;
    const float* mrow = mem + (size_t)(n0 + l) * D_ + 2 * half;  // A-matrix row
    v8f acc = {};
#pragma unroll
    for (int k0 = 0; k0 < D_; k0 += 4) {
      v2f a;  a.x  = mrow[k0];  a.y  = mrow[k0 + 1];
      v2f bm; bm.x = srow[k0];  bm.y = srow[k0 + 1];
      acc = __builtin_amdgcn_wmma_f32_16x16x4_f32(false, a, false, bm,
                                                  (short)0, acc, false, false);
    }
#pragma unroll
    for (int j = 0; j < 8; ++j)  // C layout: VGPR j -> rows j / j+8, col = l
      colsum += __cosf(SQRT_BETA * acc[j] + bia);
  }
  colsum += __shfl_xor(colsum, 16, 32);  // combine the two row-halves (same col)
  if (half == 0) atomicAdd(&T[m0 + l], SQ2M * colsum);
}

// ======================================================================
// iter: one descent step's heavy part.
//   GEMM1: H = x @ S^T (K=D), h = 2*H + bias
//   e_acc[b] += sum_m cos(h)*T[m]         (per-row shfl reduce + atomic)
//   W = sin(h)*T[m]  (LDS)
//   GEMM2: g_acc[b,:] += W @ S  (K=16 m's per wave, WMMA, LDS-reduced)
// block = 256 threads = 8 waves; wave w owns m-cols [m0, m0+16); all waves
// share the same 16 query rows b0..b0+15.
// ======================================================================
__global__ __launch_bounds__(256) void dam_iter_kernel(
    const float* __restrict__ x, const float* __restrict__ S,
    const float* __restrict__ bias, const float* __restrict__ T,
    float* __restrict__ e_acc, float* __restrict__ g_acc) {
  __shared__ float g_s[16][D_];      // 8 KB: per-block gradient tile
  __shared__ float w_s[8][16][18];   // per-wave 16x16 W tile (padded stride)

  const int tid  = threadIdx.x;
  const int wave = tid >> 5;
  const int lane = tid & 31;
  const int l    = lane & 15;
  const int half = lane >> 4;
  const int m0   = blockIdx.x * 128 + wave * 16;
  const int b0   = blockIdx.y * 16;

  for (int i = tid; i < 16 * D_; i += 256) (&g_s[0][0])[i] = 0.0f;

  // ---------------- GEMM1: acc[j] = x[b0+row] . S[m0+col] ----------------
  const float* xrow = x + (size_t)(b0 + l) * D_ + 2 * half;
  const float* srow = S + (size_t)(m0 + l) * D_ + 2 * half;
  v8f acc = {};
#pragma unroll
  for (int k0 = 0; k0 < D_; k0 += 4) {
    v2f a;  a.x  = xrow[k0];  a.y  = xrow[k0 + 1];
    v2f bm; bm.x = srow[k0];  bm.y = srow[k0 + 1];
    acc = __builtin_amdgcn_wmma_f32_16x16x4_f32(false, a, false, bm,
                                                (short)0, acc, false, false);
  }

  // ------------- nonlinearity, e-reduction, W into LDS -------------
  const float bia = bias[m0 + l];
  const float tm  = T[m0 + l];
#pragma unroll
  for (int j = 0; j < 8; ++j) {
    const int   row = j + 8 * half;           // query row within tile
    const float h   = SQRT_BETA * acc[j] + bia;
    float cv = __cosf(h);
    float sv = __sinf(h);
    w_s[wave][row][l] = sv * tm;
    float ec = cv * tm;
    ec += __shfl_xor(ec, 1, 32);
    ec += __shfl_xor(ec, 2, 32);
    ec += __shfl_xor(ec, 4, 32);
    ec += __shfl_xor(ec, 8, 32);              // sum over 16 cols in this half
    if (l == 0) atomicAdd(&e_acc[b0 + row], ec);
  }
  __syncthreads();

  // ---------------- GEMM2: G = W(16x16) @ S_tile(16x128) ----------------
#pragma unroll
  for (int nt = 0; nt < 8; ++nt) {
    const int n0 = nt * 16;
    v8f g = {};
#pragma unroll
    for (int s = 0; s < 4; ++s) {
      v2f a;
      a.x = w_s[wave][l][4 * s + 2 * half];
      a.y = w_s[wave][l][4 * s + 2 * half + 1];
      const float* sr = S + (size_t)(m0 + 4 * s + 2 * half) * D_ + n0 + l;
      v2f bm; bm.x = sr[0]; bm.y = sr[D_];
      g = __builtin_amdgcn_wmma_f32_16x16x4_f32(false, a, false, bm,
                                                (short)0, g, false, false);
    }
#pragma unroll
    for (int j = 0; j < 8; ++j)
      atomicAdd(&g_s[j + 8 * half][n0 + l], g[j]);  // LDS reduce across waves
  }
  __syncthreads();

  for (int i = tid; i < 16 * D_; i += 256) {
    const int row = i >> 7, col = i & (D_ - 1);
    atomicAdd(&g_acc[(size_t)(b0 + row) * D_ + col], g_s[row][col]);
  }
}

// ======================================================================
// update: E = -(1/4) log(max(e,eps)); x -= alpha * dE/dx; re-zero accums.
// clip gradient: zero when e_raw <= eps (matches jnp.clip's grad).
// ======================================================================
__global__ __launch_bounds__(128) void dam_update_kernel(
    float* __restrict__ x, float* __restrict__ e_acc, float* __restrict__ g_acc,
    float* __restrict__ energies, int step) {
  const int b = blockIdx.x;
  const int d = threadIdx.x;
  const float e_raw = SQ2M * e_acc[b];
  __syncthreads();  // everyone reads e_acc before thread 0 re-zeros it
  if (d == 0) {
    energies[step * B_ + b] = -0.25f * __logf(fmaxf(e_raw, EPS_));
    e_acc[b] = 0.0f;
  }
  // dE/dx = (sqrt(2/M)*sqrt(beta)/(beta*e)) * g_acc ; sqrt(beta)/beta = 0.5
  const float scale = (e_raw > EPS_) ? (ALPHA_ * 0.5f * SQ2M / e_raw) : 0.0f;
  const int idx = b * D_ + d;
  x[idx] -= scale * g_acc[idx];
  g_acc[idx] = 0.0f;
}

// ======================================================================
extern "C" void kernel_launch(void* const* d_in, const int* in_sizes, int n_in,
                              void* d_out, int out_size, void* d_ws, size_t ws_size,
                              hipStream_t stream) {
  const float* q    = (const float*)d_in[0];  // [B, D]
  const float* mem  = (const float*)d_in[1];  // [N, D]
  const float* S    = (const float*)d_in[2];  // [M, D]
  const float* bias = (const float*)d_in[3];  // [M]
  // d_in[4] = depth (device scalar) -> compile-time DEPTH_

  float* out      = (float*)d_out;
  float* x        = out;              // [B, D]  (iterated in place)
  float* energies = out + B_ * D_;    // [DEPTH, B]

  float* wsf   = (float*)d_ws;
  float* T     = wsf;                 // [M]
  float* e_acc = wsf + M_;            // [B]
  float* g_acc = wsf + M_ + B_;       // [B, D]

  dam_init_kernel<<<(WS_FLOATS + 255) / 256, 256, 0, stream>>>(q, x, wsf);
  dam_memT_kernel<<<dim3(M_ / 128, 80), 256, 0, stream>>>(mem, S, bias, T);
  for (int step = 0; step < DEPTH_; ++step) {
    dam_iter_kernel<<<dim3(M_ / 128, B_ / 16), 256, 0, stream>>>(
        x, S, bias, T, e_acc, g_acc);
    dam_update_kernel<<<B_, 128, 0, stream>>>(x, e_acc, g_acc, energies, step);
  }
}